// SetUpConv_70325794505114
// MI455X (gfx1250) — compile-verified
//
#include <hip/hip_runtime.h>
#include <hip/hip_bf16.h>

typedef __attribute__((ext_vector_type(2))) float v2f;
typedef __attribute__((ext_vector_type(8))) float v8f;

#define B_    8
#define N1_   8192
#define N2_   2048
#define KNN_K 16

// ---------------------------------------------------------------------------
// Kernel 1: KNN  — one thread per query, xyz2 (+ norms) staged in LDS.
// Register-resident sorted top-16 (fully unrolled -> promoted to VGPRs).
// Distance uses the reference's  |q|^2 + |p|^2 - 2 q.p  formula.
// ---------------------------------------------------------------------------
__global__ __launch_bounds__(256) void knn_kernel(
    const float* __restrict__ xyz1, const float* __restrict__ xyz2,
    int* __restrict__ idxbuf) {
  __shared__ __attribute__((aligned(16))) float sx[N2_], sy[N2_], sz[N2_], sn[N2_];
  const int tid = threadIdx.x;
  const int b = blockIdx.x >> 5;            // 32 blocks per batch
  const int qbase = (blockIdx.x & 31) * 256;

  const float* x2 = xyz2 + (size_t)b * N2_ * 3;
  for (int p = tid; p < N2_; p += 256) {
    float x = x2[p * 3 + 0], y = x2[p * 3 + 1], z = x2[p * 3 + 2];
    sx[p] = x; sy[p] = y; sz[p] = z;
    sn[p] = x * x + y * y + z * z;
  }
  __syncthreads();

  const int q = qbase + tid;
  const float* qp = xyz1 + ((size_t)b * N1_ + q) * 3;
  const float qx = qp[0], qy = qp[1], qz = qp[2];
  const float qn = qx * qx + qy * qy + qz * qz;

  float d[KNN_K];
  int   id[KNN_K];
#pragma unroll
  for (int j = 0; j < KNN_K; ++j) { d[j] = 3.4028235e38f; id[j] = 0; }

  for (int i = 0; i < N2_; ++i) {
    float dist = qn + sn[i] - 2.0f * (qx * sx[i] + qy * sy[i] + qz * sz[i]);
    if (dist < d[KNN_K - 1]) {
      d[KNN_K - 1] = dist; id[KNN_K - 1] = i;
#pragma unroll
      for (int j = KNN_K - 2; j >= 0; --j) {
        if (d[j + 1] < d[j]) {
          float td = d[j]; d[j] = d[j + 1]; d[j + 1] = td;
          int   ti = id[j]; id[j] = id[j + 1]; id[j + 1] = ti;
        }
      }
    }
  }
  int* o = idxbuf + ((size_t)b * N1_ + q) * KNN_K;
#pragma unroll
  for (int j = 0; j < KNN_K; ++j) o[j] = id[j];
}

// ---------------------------------------------------------------------------
// Kernel 2: gather + MLP1 (67->64->64, ReLU) + max-pool over 16 neighbors.
// One wave = one query tile: A is 16(neighbors) x 68(padded channels).
// WMMA f32 16x16x4; weights transposed in LDS with stride 68 so both A and B
// fragments are conflict-free ds_load_b64 in the documented lane layout.
// ---------------------------------------------------------------------------
__global__ __launch_bounds__(256) void mlp1_kernel(
    const float* __restrict__ xyz1, const float* __restrict__ xyz2,
    const float* __restrict__ feat2,
    const float* __restrict__ W1_0, const float* __restrict__ b1_0,
    const float* __restrict__ W1_1, const float* __restrict__ b1_1,
    const int* __restrict__ idxbuf, float* __restrict__ pooled) {
  __shared__ __attribute__((aligned(16))) float w0t[64][68];   // w0t[n][k] = W1_0[k][n]
  __shared__ __attribute__((aligned(16))) float w1t[64][68];   // w1t[n][k] = W1_1[k][n]
  __shared__ __attribute__((aligned(16))) float sb0[64], sb1[64];
  __shared__ __attribute__((aligned(16))) float Abuf[8][16][68]; // per-wave A staging
  __shared__ __attribute__((aligned(16))) float Act[8][16][68];  // per-wave layer-1 acts

  const int tid = threadIdx.x;
  for (int e = tid; e < 67 * 64; e += 256) { int k = e >> 6, n = e & 63; w0t[n][k] = W1_0[e]; }
  for (int e = tid; e < 64 * 64; e += 256) { int k = e >> 6, n = e & 63; w1t[n][k] = W1_1[e]; }
  if (tid < 64) { sb0[tid] = b1_0[tid]; sb1[tid] = b1_1[tid]; w0t[tid][67] = 0.0f; }
  __syncthreads();

  const int lane = tid & 31;
  const int w    = tid >> 5;
  const int arow = lane & 15;         // A row / B column within tile
  const int koff = (lane >> 4) << 1;  // hi half-wave handles K+2,K+3
  const int mbase = (lane >> 4) << 3; // C/D: hi half-wave holds rows 8..15

  for (int qi = 0; qi < 8; ++qi) {
    const int q  = blockIdx.x * 64 + w * 8 + qi;
    const int b  = q >> 13;
    const int n1 = q & (N1_ - 1);
    const int* idq = idxbuf + (size_t)q * KNN_K;

    // --- stage A: gather 16 feature rows (64f each) + xyz diff + zero pad ---
    {
      const int r  = lane >> 1;
      const int nb = idq[r];
      const float4* src = (const float4*)(feat2 + ((size_t)b * N2_ + nb) * 64);
      const int cbase = (lane & 1) * 8;
#pragma unroll
      for (int j = 0; j < 8; ++j) {
        float4 v = src[cbase + j];
        *(float4*)&Abuf[w][r][(cbase + j) * 4] = v;
      }
    }
    if (lane < 16) {
      const int nb = idq[lane];
      const float* p  = xyz2 + ((size_t)b * N2_ + nb) * 3;
      const float* qp = xyz1 + ((size_t)b * N1_ + n1) * 3;
      float4 v; v.x = p[0] - qp[0]; v.y = p[1] - qp[1]; v.z = p[2] - qp[2]; v.w = 0.0f;
      *(float4*)&Abuf[w][lane][64] = v;
    }
    __builtin_amdgcn_wave_barrier();

    // --- layer 0: [16x68] x [68x64] + b, ReLU -> Act ---
    for (int nt = 0; nt < 4; ++nt) {
      const int coln = nt * 16 + arow;
      const float bias = sb0[coln];
      v8f acc;
#pragma unroll
      for (int r = 0; r < 8; ++r) acc[r] = bias;
#pragma unroll
      for (int ks = 0; ks < 17; ++ks) {
        const int k = ks * 4 + koff;
        v2f a  = *(const v2f*)&Abuf[w][arow][k];
        v2f bb = *(const v2f*)&w0t[coln][k];
        acc = __builtin_amdgcn_wmma_f32_16x16x4_f32(false, a, false, bb,
                                                    (short)0, acc, false, false);
      }
#pragma unroll
      for (int r = 0; r < 8; ++r)
        Act[w][mbase + r][coln] = fmaxf(acc[r], 0.0f);
    }
    __builtin_amdgcn_wave_barrier();

    // --- layer 1: [16x64] x [64x64] + b, ReLU, max over 16 rows ---
    for (int nt = 0; nt < 4; ++nt) {
      const int coln = nt * 16 + arow;
      const float bias = sb1[coln];
      v8f acc;
#pragma unroll
      for (int r = 0; r < 8; ++r) acc[r] = bias;
#pragma unroll
      for (int ks = 0; ks < 16; ++ks) {
        const int k = ks * 4 + koff;
        v2f a  = *(const v2f*)&Act[w][arow][k];
        v2f bb = *(const v2f*)&w1t[coln][k];
        acc = __builtin_amdgcn_wmma_f32_16x16x4_f32(false, a, false, bb,
                                                    (short)0, acc, false, false);
      }
      float mx = acc[0];
#pragma unroll
      for (int r = 1; r < 8; ++r) mx = fmaxf(mx, acc[r]);
      mx = fmaxf(mx, __shfl_xor(mx, 16, 32));  // combine rows 0-7 with 8-15
      mx = fmaxf(mx, 0.0f);                    // ReLU (monotone, commutes w/ max)
      if (lane < 16) pooled[(size_t)q * 64 + coln] = mx;
    }
    __builtin_amdgcn_wave_barrier();
  }
}

// ---------------------------------------------------------------------------
// Kernel 3: concat(pooled, feat1) -> MLP2 128->128 + ReLU.
// Plain GEMM, M=65536 in 16-row tiles per wave, K=128, N=128.
// ---------------------------------------------------------------------------
__global__ __launch_bounds__(256) void mlp2_kernel(
    const float* __restrict__ pooled, const float* __restrict__ feat1,
    const float* __restrict__ W2_0, const float* __restrict__ b2_0,
    float* __restrict__ out) {
  __shared__ __attribute__((aligned(16))) float w2t[128][132];  // w2t[n][k]
  __shared__ __attribute__((aligned(16))) float sb2[128];
  __shared__ __attribute__((aligned(16))) float A2[8][16][132];

  const int tid = threadIdx.x;
  for (int e = tid; e < 128 * 128; e += 256) { int k = e >> 7, n = e & 127; w2t[n][k] = W2_0[e]; }
  if (tid < 128) sb2[tid] = b2_0[tid];
  __syncthreads();

  const int lane = tid & 31;
  const int w    = tid >> 5;
  const int arow = lane & 15;
  const int koff = (lane >> 4) << 1;
  const int mbase = (lane >> 4) << 3;

  const int tile = blockIdx.x * 8 + w;
  const int m0 = tile * 16;

  // stage A: rows m0..m0+15, cols 0-63 = pooled, 64-127 = feat1 (skip)
  {
    const int r = lane >> 1;
    const int half = lane & 1;
    const float* srcf = (half ? feat1 : pooled) + (size_t)(m0 + r) * 64;
    const float4* src = (const float4*)srcf;
#pragma unroll
    for (int j = 0; j < 16; ++j) {
      float4 v = src[j];
      *(float4*)&A2[w][r][half * 64 + j * 4] = v;
    }
  }
  __builtin_amdgcn_wave_barrier();

  for (int nt = 0; nt < 8; ++nt) {
    const int coln = nt * 16 + arow;
    const float bias = sb2[coln];
    v8f acc;
#pragma unroll
    for (int r = 0; r < 8; ++r) acc[r] = bias;
#pragma unroll
    for (int ks = 0; ks < 32; ++ks) {
      const int k = ks * 4 + koff;
      v2f a  = *(const v2f*)&A2[w][arow][k];
      v2f bb = *(const v2f*)&w2t[coln][k];
      acc = __builtin_amdgcn_wmma_f32_16x16x4_f32(false, a, false, bb,
                                                  (short)0, acc, false, false);
    }
#pragma unroll
    for (int r = 0; r < 8; ++r)
      out[(size_t)(m0 + mbase + r) * 128 + coln] = fmaxf(acc[r], 0.0f);
  }
}

// ---------------------------------------------------------------------------
extern "C" void kernel_launch(void* const* d_in, const int* in_sizes, int n_in,
                              void* d_out, int out_size, void* d_ws, size_t ws_size,
                              hipStream_t stream) {
  const float* xyz1  = (const float*)d_in[0];
  const float* feat1 = (const float*)d_in[1];
  const float* xyz2  = (const float*)d_in[2];
  const float* feat2 = (const float*)d_in[3];
  const float* W1_0  = (const float*)d_in[4];
  const float* b1_0  = (const float*)d_in[5];
  const float* W1_1  = (const float*)d_in[6];
  const float* b1_1  = (const float*)d_in[7];
  const float* W2_0  = (const float*)d_in[8];
  const float* b2_0  = (const float*)d_in[9];
  float* out = (float*)d_out;

  int*   idxbuf = (int*)d_ws;                                   // 65536*16*4  = 4 MB
  float* pooled = (float*)((char*)d_ws +
                           (size_t)B_ * N1_ * KNN_K * sizeof(int)); // 65536*64*4 = 16 MB

  knn_kernel<<<B_ * (N1_ / 256), 256, 0, stream>>>(xyz1, xyz2, idxbuf);
  mlp1_kernel<<<(B_ * N1_) / 64, 256, 0, stream>>>(xyz1, xyz2, feat2,
                                                   W1_0, b1_0, W1_1, b1_1,
                                                   idxbuf, pooled);
  mlp2_kernel<<<(B_ * N1_) / (16 * 8), 256, 0, stream>>>(pooled, feat1,
                                                         W2_0, b2_0, out);
}